// SpringMassSystem_37495064494763
// MI455X (gfx1250) — compile-verified
//
#include <hip/hip_runtime.h>

#define DT          5e-05f
#define NSUB        1000
#define SPRING_Y    30000.0f
#define DASHPOT     100.0f
// expf(-DT * DRAG_DAMPING) = exp(-5e-5), computed in double, rounded to f32
#define DRAG_MUL    0.99995000124999791f

#define N_VERT      50000
#define N_SPRING    1600000
#define TILE        256
#define SPRING_BLOCKS 1250   // 6250 tiles / 1250 blocks = 5 tiles each

// ---- CDNA5 async global->LDS copy (ASYNCcnt-tracked DMA into LDS) ----------
__device__ __forceinline__ void async_ld_b64(unsigned lds_addr, const void* gptr) {
    asm volatile("global_load_async_to_lds_b64 %0, %1, off"
                 :: "v"(lds_addr), "v"((unsigned long long)gptr) : "memory");
}
__device__ __forceinline__ void async_ld_b32(unsigned lds_addr, const void* gptr) {
    asm volatile("global_load_async_to_lds_b32 %0, %1, off"
                 :: "v"(lds_addr), "v"((unsigned long long)gptr) : "memory");
}
#define WAIT_ASYNC(n) asm volatile("s_wait_asynccnt %0" :: "i"(n) : "memory")

__device__ __forceinline__ unsigned lds_off(const void* p) {
    // flat shared address: LDS offset is addr[31:0] (ISA 10.2 aperture rules)
    return (unsigned)(unsigned long long)p;
}

// ---------------------------------------------------------------------------
// Init: copy x from input, zero v and force accumulator (re-done every call)
// ---------------------------------------------------------------------------
__global__ void sms_init_kernel(const float* __restrict__ x0,
                                float* __restrict__ xw,
                                float* __restrict__ vw,
                                float* __restrict__ fw, int n) {
    int i = blockIdx.x * blockDim.x + threadIdx.x;
    if (i < n) { xw[i] = x0[i]; vw[i] = 0.0f; fw[i] = 0.0f; }
}

// ---------------------------------------------------------------------------
// Spring pass: per-spring force (spring + dashpot merged), scatter-add ±f.
// Spring records are streamed via double-buffered async copies into LDS.
// Each thread stages/reads only its own LDS slot -> per-wave ASYNCcnt waits,
// no block barriers.
// ---------------------------------------------------------------------------
__global__ void __launch_bounds__(TILE)
sms_spring_kernel(const float* __restrict__ x,
                  const float* __restrict__ v,
                  float*       __restrict__ f,
                  const int*   __restrict__ springs,   // (N_SPRING, 2)
                  const float* __restrict__ rest,      // (N_SPRING,)
                  float*       __restrict__ sf_out) {  // last substep only
    __shared__ int2  s_idx [2][TILE];
    __shared__ float s_rest[2][TILE];

    const int tid      = threadIdx.x;
    const int numTiles = N_SPRING / TILE;   // 6250, exact
    int tile = blockIdx.x;
    if (tile >= numTiles) return;

    // stage first tile
    {
        int j = tile * TILE + tid;
        async_ld_b64(lds_off(&s_idx [0][tid]), springs + 2 * j);
        async_ld_b32(lds_off(&s_rest[0][tid]), rest + j);
    }

    int buf = 0;
    for (; tile < numTiles; tile += gridDim.x) {
        int nextTile = tile + gridDim.x;
        if (nextTile < numTiles) {
            int j = nextTile * TILE + tid;
            async_ld_b64(lds_off(&s_idx [buf ^ 1][tid]), springs + 2 * j);
            async_ld_b32(lds_off(&s_rest[buf ^ 1][tid]), rest + j);
            WAIT_ASYNC(2);   // current tile done; next tile's 2 ops may remain
        } else {
            WAIT_ASYNC(0);
        }

        int2  sp = s_idx [buf][tid];
        float r  = s_rest[buf][tid];
        int b1 = 3 * sp.x;
        int b2 = 3 * sp.y;

        float x1x = x[b1], x1y = x[b1 + 1], x1z = x[b1 + 2];
        float x2x = x[b2], x2y = x[b2 + 1], x2z = x[b2 + 2];
        float dx = x2x - x1x, dy = x2y - x1y, dz = x2z - x1z;
        float len  = sqrtf(dx * dx + dy * dy + dz * dz);
        float inv  = 1.0f / len;
        float ux = dx * inv, uy = dy * inv, uz = dz * inv;

        float v1x = v[b1], v1y = v[b1 + 1], v1z = v[b1 + 2];
        float v2x = v[b2], v2y = v[b2 + 1], v2z = v[b2 + 2];
        float vrel = (v2x - v1x) * ux + (v2y - v1y) * uy + (v2z - v1z) * uz;

        float fs   = SPRING_Y * (len / r - 1.0f);   // spring term only (output)
        float coef = fs + DASHPOT * vrel;           // merged with dashpot
        float fxv = coef * ux, fyv = coef * uy, fzv = coef * uz;

        unsafeAtomicAdd(&f[b1],     fxv);
        unsafeAtomicAdd(&f[b1 + 1], fyv);
        unsafeAtomicAdd(&f[b1 + 2], fzv);
        unsafeAtomicAdd(&f[b2],     -fxv);
        unsafeAtomicAdd(&f[b2 + 1], -fyv);
        unsafeAtomicAdd(&f[b2 + 2], -fzv);

        if (sf_out) {
            int j = tile * TILE + tid;
            sf_out[3 * j]     = fs * ux;
            sf_out[3 * j + 1] = fs * uy;
            sf_out[3 * j + 2] = fs * uz;
        }
        buf ^= 1;
    }
}

// ---------------------------------------------------------------------------
// Vertex pass: gravity, integrate, drag, floor clamp; zero f for next substep
// ---------------------------------------------------------------------------
__global__ void sms_vertex_kernel(float* __restrict__ x,
                                  float* __restrict__ v,
                                  float* __restrict__ f,
                                  const float* __restrict__ m,
                                  float* __restrict__ xout) {
    int i = blockIdx.x * blockDim.x + threadIdx.x;
    if (i >= N_VERT) return;
    int b = 3 * i;
    float mi = m[i];
    float fx = f[b], fy = f[b + 1], fz = f[b + 2] - 9.8f * mi;
    f[b] = 0.0f; f[b + 1] = 0.0f; f[b + 2] = 0.0f;

    float vx = (v[b]     + DT * fx / mi) * DRAG_MUL;
    float vy = (v[b + 1] + DT * fy / mi) * DRAG_MUL;
    float vz = (v[b + 2] + DT * fz / mi) * DRAG_MUL;

    float nx = x[b]     + DT * vx;
    float ny = x[b + 1] + DT * vy;
    float nz = x[b + 2] + DT * vz;
    nz = fmaxf(nz, 0.0f);
    if (nz == 0.0f) vz = 0.0f;

    x[b] = nx; x[b + 1] = ny; x[b + 2] = nz;
    v[b] = vx; v[b + 1] = vy; v[b + 2] = vz;
    if (xout) { xout[b] = nx; xout[b + 1] = ny; xout[b + 2] = nz; }
}

// ---------------------------------------------------------------------------
extern "C" void kernel_launch(void* const* d_in, const int* in_sizes, int n_in,
                              void* d_out, int out_size, void* d_ws, size_t ws_size,
                              hipStream_t stream) {
    const float* x0      = (const float*)d_in[0];   // (N_VERT, 3)
    const int*   springs = (const int*)  d_in[1];   // (N_SPRING, 2)
    const float* rest    = (const float*)d_in[2];   // (N_SPRING,)
    const float* masses  = (const float*)d_in[3];   // (N_VERT,)
    float* out = (float*)d_out;                     // [x | spring_forces]

    float* xw = (float*)d_ws;
    float* vw = xw + 3 * N_VERT;
    float* fw = vw + 3 * N_VERT;

    const int nState = 3 * N_VERT;
    sms_init_kernel<<<(nState + 255) / 256, 256, 0, stream>>>(x0, xw, vw, fw, nState);

    for (int s = 0; s < NSUB; ++s) {
        const bool last = (s == NSUB - 1);
        sms_spring_kernel<<<SPRING_BLOCKS, TILE, 0, stream>>>(
            xw, vw, fw, springs, rest, last ? (out + 3 * N_VERT) : nullptr);
        sms_vertex_kernel<<<(N_VERT + 255) / 256, 256, 0, stream>>>(
            xw, vw, fw, masses, last ? out : nullptr);
    }
}